// GCN_53206054863534
// MI455X (gfx1250) — compile-verified
//
#include <hip/hip_runtime.h>
#include <hip/hip_bf16.h>
#include <math.h>

// GCN forward on MI455X (gfx1250, wave32).
// Strategy:
//  * Build CSR (by dst, self-loop in slot 0 of each row) once per call:
//      deg -> dinv -> exclusive scan (3-kernel) -> CSR fill.
//  * Per layer: dense GEMM X@W via V_WMMA_F32_16X16X4_F32 (fp32 WMMA, full
//    precision; GEMMs are bandwidth-bound so fp32 costs nothing). W is staged
//    in LDS in *fragment order* (K-pairs adjacent per column) so each B
//    fragment is a single ds_load_b64. X rows are streamed with
//    global_prefetch_b8 ahead of the WMMA cadence.
//  * CSR gather-aggregate (atomic-free; h matrices are <=25.6MB so fully
//    L2-resident in the 192MB L2) fused with bias/ReLU and the extra outputs
//    (x2, x5, x8). Final layer fuses log_softmax.

typedef __attribute__((ext_vector_type(2))) float v2f;
typedef __attribute__((ext_vector_type(8))) float v8f;

#define TPB 256

// ---------------- CSR construction ----------------

__global__ void k_init_deg(int* deg, int n) {
  int i = blockIdx.x * blockDim.x + threadIdx.x;
  if (i < n) deg[i] = 1;  // self-loop
}

__global__ void k_count_deg(const int* __restrict__ edge, int* deg, int e) {
  int i = blockIdx.x * blockDim.x + threadIdx.x;
  if (i < e) atomicAdd(&deg[edge[e + i]], 1);  // dst row
}

__global__ void k_dinv(const int* __restrict__ deg, float* dinv, int n) {
  int i = blockIdx.x * blockDim.x + threadIdx.x;
  if (i < n) dinv[i] = rsqrtf((float)deg[i]);  // deg >= 1 always
}

__global__ void k_scan_block(const int* __restrict__ deg, int* rowptr,
                             int* bsum, int n) {
  __shared__ int s[TPB];
  int tid = threadIdx.x;
  int i = blockIdx.x * TPB + tid;
  int v = (i < n) ? deg[i] : 0;
  s[tid] = v;
  __syncthreads();
  for (int off = 1; off < TPB; off <<= 1) {
    int t = (tid >= off) ? s[tid - off] : 0;
    __syncthreads();
    s[tid] += t;
    __syncthreads();
  }
  if (i < n) rowptr[i] = s[tid] - v;  // exclusive within block
  if (tid == TPB - 1) bsum[blockIdx.x] = s[tid];
}

__global__ void k_scan_bsums(int* bsum, int nb) {
  __shared__ int s[512];
  int tid = threadIdx.x;
  int v = (tid < nb) ? bsum[tid] : 0;
  s[tid] = v;
  __syncthreads();
  for (int off = 1; off < 512; off <<= 1) {
    int t = (tid >= off) ? s[tid - off] : 0;
    __syncthreads();
    s[tid] += t;
    __syncthreads();
  }
  if (tid < nb) bsum[tid] = s[tid] - v;  // exclusive
}

__global__ void k_scan_add(int* rowptr, const int* __restrict__ bsum, int n,
                           int total) {
  int i = blockIdx.x * blockDim.x + threadIdx.x;
  if (i < n) rowptr[i] += bsum[blockIdx.x];
  if (i == 0) rowptr[n] = total;
}

// self-loop occupies deterministic slot rowptr[i]; cursor starts just after
__global__ void k_init_csr(const int* __restrict__ rowptr,
                           const float* __restrict__ dinv, int* cursor,
                           int* ccol, float* cval, int n) {
  int i = blockIdx.x * blockDim.x + threadIdx.x;
  if (i < n) {
    int p = rowptr[i];
    cursor[i] = p + 1;
    ccol[p] = i;
    float d = dinv[i];
    cval[p] = d * d;
  }
}

__global__ void k_fill_edges(const int* __restrict__ edge,
                             const float* __restrict__ dinv, int* cursor,
                             int* ccol, float* cval, int e) {
  int i = blockIdx.x * blockDim.x + threadIdx.x;
  if (i < e) {
    int s = edge[i];
    int d = edge[e + i];
    int slot = atomicAdd(&cursor[d], 1);
    ccol[slot] = s;
    cval[slot] = dinv[s] * dinv[d];
  }
}

// ---------------- Dense GEMM via fp32 WMMA ----------------
// H[N x F] = X[N x K] @ W[K x F].  F padded to Fpad (mult of 16) with zeros
// in LDS.  LDS holds W in fragment order: pair-row p (= K pair {2p,2p+1}),
// column c stored at sW[p*2*Fpad + 2*c + {0,1}] so a B fragment is one
// 8B-aligned ds_load_b64.  8 waves/WG; wave w -> (subRow = w/nCT,
// colTile = w%nCT).  K is a multiple of 4 for every layer.
__global__ __launch_bounds__(TPB) void k_gemm_wmma(
    const float* __restrict__ X, const float* __restrict__ W,
    float* __restrict__ H, int n, int K, int F, int Fpad, int nCT) {
  extern __shared__ float sW[];
  int Fp2 = Fpad * 2;
  // stage W into LDS (zero-pad cols), fragment-order layout
  int tot = (K >> 1) * Fp2;
  for (int idx = threadIdx.x; idx < tot; idx += TPB) {
    int p = idx / Fp2;
    int rem = idx - p * Fp2;
    int c = rem >> 1;
    int k = (p << 1) + (rem & 1);
    sW[idx] = (c < F) ? W[k * F + c] : 0.0f;
  }
  __syncthreads();

  int wave = threadIdx.x >> 5;
  int lane = threadIdx.x & 31;
  int laneLo = lane & 15;
  int koff = (lane >> 4) << 1;  // lanes 0-15 -> K {k,k+1}; 16-31 -> {k+2,k+3}
  int rowsPerWG = (8 / nCT) * 16;
  int subRow = wave / nCT;
  int colTile = wave - subRow * nCT;
  int rowBase = blockIdx.x * rowsPerWG + subRow * 16;
  int row = rowBase + laneLo;
  int rowC = row < n ? row : n - 1;  // clamp OOB loads, stores are guarded
  int col = colTile * 16 + laneLo;

  const float* xp = X + (size_t)rowC * K + koff;
  const float* bp = sW + (koff >> 1) * Fp2 + (col << 1);
  v8f c = {};
  for (int k = 0; k < K; k += 4) {
    if ((k & 31) == 0)  // one speculative prefetch per 128B cacheline, 256B ahead
      __builtin_prefetch(xp + k + 64, 0, 1);
    v2f a = *reinterpret_cast<const v2f*>(xp + k);            // global b64
    v2f b = *reinterpret_cast<const v2f*>(bp + (k >> 1) * Fp2);  // ds b64
    c = __builtin_amdgcn_wmma_f32_16x16x4_f32(false, a, false, b, (short)0, c,
                                              false, false);
  }
  if (col < F) {
    int rhalf = (lane >> 4) * 8;  // VGPR r: lanes0-15 M=r, lanes16-31 M=r+8
    for (int r = 0; r < 8; ++r) {
      int orow = rowBase + rhalf + r;
      if (orow < n) H[(size_t)orow * F + col] = c[r];
    }
  }
}

// ---------------- CSR aggregation (fused bias/relu/outputs) ----------------
// thread = (node, feature); F consecutive threads gather the same h row ->
// fully coalesced 4*F-byte L2-resident reads per edge.
__global__ void k_aggregate(const float* __restrict__ H,
                            const int* __restrict__ rowptr,
                            const int* __restrict__ ccol,
                            const float* __restrict__ cval,
                            const float* __restrict__ bias,
                            float* __restrict__ out_act,
                            float* __restrict__ out_raw, int n, int F,
                            int doRelu) {
  int t = blockIdx.x * blockDim.x + threadIdx.x;
  int node = t / F;
  if (node >= n) return;
  int f = t - node * F;
  int e0 = rowptr[node], e1 = rowptr[node + 1];
  float acc = 0.0f;
  for (int e = e0; e < e1; ++e) {
    acc += cval[e] * H[(size_t)ccol[e] * F + f];
  }
  float v = acc + bias[f];
  if (out_raw) out_raw[(size_t)node * F + f] = v;
  if (doRelu) v = fmaxf(v, 0.0f);
  out_act[(size_t)node * F + f] = v;
}

// final layer: aggregate 3 features per node + log_softmax
__global__ void k_final(const float* __restrict__ H,
                        const int* __restrict__ rowptr,
                        const int* __restrict__ ccol,
                        const float* __restrict__ cval,
                        const float* __restrict__ bias,
                        float* __restrict__ out, int n) {
  int node = blockIdx.x * blockDim.x + threadIdx.x;
  if (node >= n) return;
  int e0 = rowptr[node], e1 = rowptr[node + 1];
  float a0 = 0.f, a1 = 0.f, a2 = 0.f;
  for (int e = e0; e < e1; ++e) {
    float w = cval[e];
    const float* h = H + (size_t)ccol[e] * 3;
    a0 += w * h[0];
    a1 += w * h[1];
    a2 += w * h[2];
  }
  a0 += bias[0];
  a1 += bias[1];
  a2 += bias[2];
  float m = fmaxf(a0, fmaxf(a1, a2));
  float l = logf(expf(a0 - m) + expf(a1 - m) + expf(a2 - m));
  float* o = out + (size_t)node * 3;
  o[0] = a0 - m - l;
  o[1] = a1 - m - l;
  o[2] = a2 - m - l;
}

// ---------------- host launch ----------------

extern "C" void kernel_launch(void* const* d_in, const int* in_sizes, int n_in,
                              void* d_out, int out_size, void* d_ws,
                              size_t ws_size, hipStream_t stream) {
  const int K0 = 500;
  const float* x = (const float*)d_in[0];
  const int* edge = (const int*)d_in[1];
  const float* W[5] = {(const float*)d_in[2], (const float*)d_in[4],
                       (const float*)d_in[6], (const float*)d_in[8],
                       (const float*)d_in[10]};
  const float* B[5] = {(const float*)d_in[3], (const float*)d_in[5],
                       (const float*)d_in[7], (const float*)d_in[9],
                       (const float*)d_in[11]};
  const int N = in_sizes[0] / K0;  // 100000
  const int E = in_sizes[1] / 2;   // 1600000
  const int NE = N + E;

  // workspace carve-up (256B aligned)
  char* w = (char*)d_ws;
  size_t off = 0;
  auto alloc = [&](size_t bytes) {
    void* p = w + off;
    off = (off + bytes + 255) & ~(size_t)255;
    return p;
  };
  int* deg = (int*)alloc((size_t)N * 4);
  float* dinv = (float*)alloc((size_t)N * 4);
  int* rowptr = (int*)alloc((size_t)(N + 1) * 4);
  int* cursor = (int*)alloc((size_t)N * 4);
  int* bsum = (int*)alloc(512 * 4);
  int* ccol = (int*)alloc((size_t)NE * 4);
  float* cval = (float*)alloc((size_t)NE * 4);
  float* H = (float*)alloc((size_t)N * 64 * 4);    // GEMM output, reused
  float* act = (float*)alloc((size_t)N * 32 * 4);  // x4 / x6, reused
  (void)ws_size;

  // output regions: [log_softmax N*3][x2 N*64][x5 N*24][x8 N*16]
  float* out = (float*)d_out;
  float* o_ls = out;
  float* o_x2 = out + (size_t)N * 3;
  float* o_x5 = out + (size_t)N * 67;
  float* o_x8 = out + (size_t)N * 91;

  int nbN = (N + TPB - 1) / TPB;  // 391
  int nbE = (E + TPB - 1) / TPB;  // 6250

  // ---- CSR build ----
  k_init_deg<<<nbN, TPB, 0, stream>>>(deg, N);
  k_count_deg<<<nbE, TPB, 0, stream>>>(edge, deg, E);
  k_dinv<<<nbN, TPB, 0, stream>>>(deg, dinv, N);
  k_scan_block<<<nbN, TPB, 0, stream>>>(deg, rowptr, bsum, N);
  k_scan_bsums<<<1, 512, 0, stream>>>(bsum, nbN);
  k_scan_add<<<nbN, TPB, 0, stream>>>(rowptr, bsum, N, NE);
  k_init_csr<<<nbN, TPB, 0, stream>>>(rowptr, dinv, cursor, ccol, cval, N);
  k_fill_edges<<<nbE, TPB, 0, stream>>>(edge, dinv, cursor, ccol, cval, E);

  auto gemm = [&](const float* X, const float* Wt, float* Ht, int Kd, int Fd,
                  int Fpad) {
    int nCT = Fpad / 16;
    if (nCT > 4) nCT = 4;  // 8 waves: subRows x colTiles
    int rowsPerWG = (8 / nCT) * 16;
    int blocks = (N + rowsPerWG - 1) / rowsPerWG;
    size_t sh = (size_t)(Kd >> 1) * (Fpad * 2) * 4;
    k_gemm_wmma<<<blocks, TPB, sh, stream>>>(X, Wt, Ht, N, Kd, Fd, Fpad, nCT);
  };
  auto agg = [&](const float* Ht, const float* bias, float* oact, float* oraw,
                 int Fd, int relu) {
    int blocks = ((size_t)N * Fd + TPB - 1) / TPB;
    k_aggregate<<<blocks, TPB, 0, stream>>>(Ht, rowptr, ccol, cval, bias, oact,
                                            oraw, N, Fd, relu);
  };

  // L1: x2 = relu(agg(x@W1)+b1)            -> d_out.x2 (also next input)
  gemm(x, W[0], H, 500, 64, 64);
  agg(H, B[0], o_x2, nullptr, 64, 1);
  // L2: x4 = relu(agg(x2@W2)+b2)           -> ws act
  gemm(o_x2, W[1], H, 64, 32, 32);
  agg(H, B[1], act, nullptr, 32, 1);
  // L3: x5 = agg(x4@W3)+b3 -> d_out.x5 ; x6 = relu(x5) -> ws act
  gemm(act, W[2], H, 32, 24, 32);
  agg(H, B[2], act, o_x5, 24, 1);
  // L4: x8 = agg(x6@W4)+b4 (no relu)       -> d_out.x8 (also next input)
  gemm(act, W[3], H, 24, 16, 16);
  agg(H, B[3], o_x8, nullptr, 16, 0);
  // L5: log_softmax(agg(x8@W5)+b5)         -> d_out.ls
  gemm(o_x8, W[4], H, 16, 3, 16);
  k_final<<<nbN, TPB, 0, stream>>>(H, rowptr, ccol, cval, B[4], o_ls, N);
}